// S4Conv_59459527246264
// MI455X (gfx1250) — compile-verified
//
#include <hip/hip_runtime.h>
#include <hip/hip_bf16.h>

// ---------------------------------------------------------------------------
// Reformulated S4-style linear scan for MI455X (gfx1250, wave32, WMMA):
//   h_{i+1} = A1 h_i + B1 x_i  (L=262144, N=64) ; y = C1 h_L
//
// h_L = sum_i A1^{L-1-i} B1 x_i.  Split L = C*K chunks (C=4096, K=64):
//   W_j = A1^{K-1-j} B1            (1 MB table, L2-resident, shared by all blocks)
//   p_c = sum_j W_j x_{cK+j}       -> GEMM (64 x 4096 x 4096 cols) via f32 WMMA,
//                                     x-tiles double-buffered in LDS via
//                                     GLOBAL_LOAD_ASYNC_TO_LDS (ASYNCcnt)
//   h_L = 12-level log-depth tree combine with factors A1^(K*2^l)
// ---------------------------------------------------------------------------

typedef __attribute__((ext_vector_type(2))) float v2f;
typedef __attribute__((ext_vector_type(8))) float v8f;
typedef __attribute__((ext_vector_type(4))) int i32x4;

#define KCH 64    // timesteps per chunk
#define NCH 4096  // chunks (KCH*NCH == 262144)
#define LVLS 12   // log2(NCH)

#if defined(__gfx1250__) && __has_builtin(__builtin_amdgcn_global_load_async_to_lds_b128)
#define HAVE_ASYNC_LDS 1
#else
#define HAVE_ASYNC_LDS 0
#endif

__device__ __forceinline__ void async_copy16(const float* g, float* l) {
#if HAVE_ASYNC_LDS
  __builtin_amdgcn_global_load_async_to_lds_b128(
      (__attribute__((address_space(1))) i32x4*)g,
      (__attribute__((address_space(3))) i32x4*)l, 0, 0);
#else
  *(float4*)l = *(const float4*)g;
#endif
}

__device__ __forceinline__ void wait_async0() {
#if HAVE_ASYNC_LDS
#if __has_builtin(__builtin_amdgcn_s_wait_asynccnt)
  __builtin_amdgcn_s_wait_asynccnt(0);
#else
  asm volatile("s_wait_asynccnt 0" ::: "memory");
#endif
#endif
}

// ---- Phase 1: ladder[t] = A^(2^t), t = 0..17 (single workgroup) -----------
__global__ __launch_bounds__(256) void s4_powers(const float* __restrict__ A,
                                                 float* __restrict__ ladder) {
  __shared__ float cur[4096];
  __shared__ float nxt[4096];
  const int tid = threadIdx.x;
  for (int i = tid; i < 4096; i += 256) { cur[i] = A[i]; ladder[i] = A[i]; }
  __syncthreads();
  for (int t = 1; t < 18; ++t) {
    for (int i = tid; i < 4096; i += 256) {
      const int m = i >> 6, p = i & 63;
      float s = 0.f;
      for (int k = 0; k < 64; ++k) s = fmaf(cur[m * 64 + k], cur[k * 64 + p], s);
      nxt[i] = s;
    }
    __syncthreads();
    for (int i = tid; i < 4096; i += 256) {
      cur[i] = nxt[i];
      ladder[t * 4096 + i] = nxt[i];
    }
    __syncthreads();
  }
}

// ---- Phase 2: Wt[j] = A^(K-1-j) @ B, one block per j ----------------------
__global__ __launch_bounds__(256) void s4_wtable(const float* __restrict__ B,
                                                 const float* __restrict__ ladder,
                                                 float* __restrict__ Wt) {
  __shared__ float cur[4096];
  __shared__ float nxt[4096];
  __shared__ float Lm[4096];
  const int tid = threadIdx.x;
  const int j = blockIdx.x;
  const int e = (KCH - 1) - j;  // exponent, uniform per block
  for (int i = tid; i < 4096; i += 256) cur[i] = B[i];
  __syncthreads();
  for (int t = 0; t < 6; ++t) {     // KCH-1 fits in 6 bits
    if (!((e >> t) & 1)) continue;  // uniform branch: barriers stay legal
    for (int i = tid; i < 4096; i += 256) Lm[i] = ladder[t * 4096 + i];
    __syncthreads();
    for (int i = tid; i < 4096; i += 256) {
      const int m = i >> 6, p = i & 63;
      float s = 0.f;
      for (int k = 0; k < 64; ++k) s = fmaf(Lm[m * 64 + k], cur[k * 64 + p], s);
      nxt[i] = s;
    }
    __syncthreads();
    for (int i = tid; i < 4096; i += 256) cur[i] = nxt[i];
    __syncthreads();
  }
  for (int i = tid; i < 4096; i += 256) Wt[(size_t)j * 4096 + i] = cur[i];
}

// ---- Phase 3: P[c] = sum_j Wt[j] @ x[cK+j]  via V_WMMA_F32_16X16X4_F32 ----
// Block = 4 waves (128 threads); wave w owns M rows 16w..16w+15; block owns
// 16 chunk columns.  x-tiles (16 chunks x 64 feat) are async-staged into LDS
// (double buffered) and shared by all 4 waves; W rows stream from L2.
__global__ __launch_bounds__(128) void s4_wmma_gemm(const float* __restrict__ x,
                                                    const float* __restrict__ Wt,
                                                    float* __restrict__ P) {
  __shared__ float xs[2][16 * 64];  // [buf][chunk-in-tile][feature]
  const int tid     = threadIdx.x;
  const int lane    = tid & 31;
  const int wave    = tid >> 5;   // 0..3 -> M tile
  const int lane_lo = lane & 15;
  const int hi      = lane >> 4;  // 0: K cols {0,1}; 1: K cols {2,3}
  const int m_base  = wave * 16;
  const int c_base  = blockIdx.x * 16;

  // Staging map: thread copies 8 floats (2 x b128) of the 16x64 tile.
  const int sn = tid >> 3;        // chunk-in-tile 0..15
  const int sp = (tid & 7) * 8;   // feature base 0..56
  const float* gstage = x + ((size_t)(c_base + sn) * KCH) * 64 + sp;  // + j*64
  const int lds_off = sn * 64 + sp;

  // A-frag source: row (m_base+lane_lo) of W_j, cols (4q + 2*hi + {0,1})
  const float* wrow = Wt + (size_t)(m_base + lane_lo) * 64 + 2 * hi;

  // Prologue: stage tile j=0 into buffer 0.
  async_copy16(gstage, &xs[0][lds_off]);
  async_copy16(gstage + 4, &xs[0][lds_off + 4]);
  wait_async0();
  __syncthreads();

  v8f acc = {};
  for (int j = 0; j < KCH; ++j) {
    const int cb = j & 1;
    if (j + 1 < KCH) {  // issue async stage of tile j+1 into the other buffer
      const float* g = gstage + (size_t)(j + 1) * 64;
      async_copy16(g, &xs[cb ^ 1][lds_off]);
      async_copy16(g + 4, &xs[cb ^ 1][lds_off + 4]);
    }
    __builtin_prefetch(wrow + (size_t)(j + 4) * 4096, 0, 0);  // W stream ahead

    const float* wj = wrow + (size_t)j * 4096;
    const float* xb = &xs[cb][lane_lo * 64 + 2 * hi];
#pragma unroll
    for (int q = 0; q < 16; ++q) {  // 16 WMMAs cover the 64-wide inner slice
      v2f a = *(const v2f*)(wj + q * 4);  // global (L2-resident W)
      v2f b = *(const v2f*)(xb + q * 4);  // LDS (ds_load_b64)
      acc = __builtin_amdgcn_wmma_f32_16x16x4_f32(
          /*neg_a=*/false, a, /*neg_b=*/false, b,
          /*c_mod=*/(short)0, acc, /*reuse_a=*/false, /*reuse_b=*/false);
    }
    wait_async0();    // next tile resident
    __syncthreads();  // all waves done reading xs[cb]; writes visible
  }

  // D layout: VGPR r holds (M = m_base + r + 8*hi, N = lane_lo)
  float* pout = P + (size_t)(c_base + lane_lo) * 64 + m_base + 8 * hi;
#pragma unroll
  for (int r = 0; r < 8; ++r) pout[r] = acc[r];
}

// ---- Phase 4: tree combine  out[b] = Al @ in[2b] + in[2b+1] ---------------
__global__ __launch_bounds__(64) void s4_combine(const float* __restrict__ in,
                                                 float* __restrict__ out,
                                                 const float* __restrict__ Al) {
  __shared__ float h0[64];
  const int m = threadIdx.x;
  const int b = blockIdx.x;
  h0[m] = in[(size_t)(2 * b) * 64 + m];
  __syncthreads();
  float s = in[(size_t)(2 * b + 1) * 64 + m];
  for (int k = 0; k < 64; ++k) s = fmaf(Al[m * 64 + k], h0[k], s);
  out[(size_t)b * 64 + m] = s;
}

// ---- Phase 5: y = C1 @ h ---------------------------------------------------
__global__ __launch_bounds__(64) void s4_final(const float* __restrict__ C1,
                                               const float* __restrict__ h,
                                               float* __restrict__ y) {
  __shared__ float hh[64];
  const int m = threadIdx.x;
  hh[m] = h[m];
  __syncthreads();
  float s = 0.f;
  for (int k = 0; k < 64; ++k) s = fmaf(C1[m * 64 + k], hh[k], s);
  y[m] = s;
}

extern "C" void kernel_launch(void* const* d_in, const int* in_sizes, int n_in,
                              void* d_out, int out_size, void* d_ws, size_t ws_size,
                              hipStream_t stream) {
  (void)in_sizes; (void)n_in; (void)out_size; (void)ws_size;
  const float* x  = (const float*)d_in[0];  // (262144, 64)
  const float* A1 = (const float*)d_in[1];  // (64, 64)
  const float* B1 = (const float*)d_in[2];  // (64, 64)
  const float* C1 = (const float*)d_in[3];  // (64, 64)

  float* ws     = (float*)d_ws;
  float* ladder = ws;                  // 18 * 4096 floats
  float* Wt     = ladder + 18 * 4096;  // KCH * 4096 floats (1 MB)
  float* P0     = Wt + KCH * 4096;     // NCH * 64 floats
  float* P1     = P0 + NCH * 64;       // NCH * 64 floats

  s4_powers<<<1, 256, 0, stream>>>(A1, ladder);
  s4_wtable<<<KCH, 256, 0, stream>>>(B1, ladder, Wt);
  s4_wmma_gemm<<<NCH / 16, 128, 0, stream>>>(x, Wt, P0);

  float* bufs[2] = {P0, P1};
  int cnt = NCH / 2;
  for (int lvl = 0; lvl < LVLS; ++lvl) {  // combine factor = A^(KCH * 2^lvl)
    const float* in = bufs[lvl & 1];
    float* out      = bufs[(lvl & 1) ^ 1];
    s4_combine<<<cnt, 64, 0, stream>>>(in, out, ladder + (size_t)(6 + lvl) * 4096);
    cnt >>= 1;
  }
  // lvl 11 wrote into bufs[0] == P0; h_L is P0[0..63]
  s4_final<<<1, 64, 0, stream>>>(C1, P0, (float*)d_out);
}